// MambaSubBlock_70145405878728
// MI455X (gfx1250) — compile-verified
//
#include <hip/hip_runtime.h>
#include <hip/hip_bf16.h>
#include <math.h>
#include <stdint.h>

// ---------------------------------------------------------------------------
// Mamba block for MI455X (gfx1250, wave32, WMMA + async global->LDS staging).
// All GEMMs run on V_WMMA_F32_16X16X4_F32 (exact fp32 matrix math).
// ---------------------------------------------------------------------------

typedef __attribute__((ext_vector_type(2))) float v2f;
typedef __attribute__((ext_vector_type(8))) float v8f;

#define DIM      1024
#define D_STATE  16
#define D_CONV   4
#define D_INNER  2048
#define DT_RANK  64
#define B_SZ     2
#define SEQ      1024
#define ROWS     (B_SZ * SEQ)          // 2048 (M for all GEMMs)

#define BM 128                         // block M tile (8 waves x 16 rows)
#define BN 64                          // block N tile (wave computes 16x64)
#define KC 16                          // K chunk (4 WMMA k-steps)
#define APAD 20                        // KC+4: 16B-aligned rows for async b128
#define BROW 160                       // sBp row stride (dwords): mod64==32 ->
                                       // lane halves hit disjoint bank halves

__device__ __forceinline__ float silu_f(float x) {
    return x / (1.0f + __expf(-x));
}
__device__ __forceinline__ float softplus_f(float x) {
    return fmaxf(x, 0.0f) + log1pf(__expf(-fabsf(x)));
}

// low 32 bits of a flat shared pointer == wave-relative LDS byte address
__device__ __forceinline__ uint32_t lds_addr_of(const void* p) {
    return (uint32_t)(uintptr_t)p;
}
__device__ __forceinline__ void async_b128(uint32_t lds, const void* gaddr) {
    asm volatile("global_load_async_to_lds_b128 %0, %1, off"
                 :: "v"(lds), "v"(gaddr) : "memory");
}
__device__ __forceinline__ void async_b128_off16(uint32_t lds, const void* gaddr) {
    // INST_OFFSET is added to both the LDS and the global address (ISA §10.7)
    asm volatile("global_load_async_to_lds_b128 %0, %1, off offset:16"
                 :: "v"(lds), "v"(gaddr) : "memory");
}
__device__ __forceinline__ void wait_async0() {
    asm volatile("s_wait_asynccnt 0" ::: "memory");
}

// ---------------------------------------------------------------------------
// fp32 WMMA GEMM:  C[M,N] = A[M,K] @ W[K,N]   (row-major, arbitrary ld)
// EPI = 0: none   1: softplus(acc + bias[n])   2: mask[row] ? acc : 0
// 256 threads = 8 waves; wave w computes rows [m0+16w, m0+16w+16) x 64 cols.
// A: async global->LDS, double buffered.  B: K-pair-interleaved LDS layout
// (sBp[k/2][col][k&1]) so every WMMA B fragment is one contiguous b64 read.
// ---------------------------------------------------------------------------
template <int EPI>
__global__ __launch_bounds__(256) void wmma_gemm_f32(
    const float* __restrict__ A, const float* __restrict__ W,
    float* __restrict__ C,
    int M, int N, int K, int lda, int ldb, int ldc,
    const float* __restrict__ bias, const unsigned char* __restrict__ mask)
{
    __shared__ float sA[2][BM][APAD];       // 2 x 128 x 20 floats (async dest)
    __shared__ float sBp[2][KC / 2][BROW];  // 2 x 8 x 160 floats (interleaved)

    const int tid  = threadIdx.x;
    const int lane = tid & 31;
    const int wave = tid >> 5;          // 0..7 -> 16-row M strip

    const int m0 = blockIdx.y * BM;
    const int n0 = blockIdx.x * BN;

    const int lhalf = lane & 15;
    const int khalf = (lane < 16) ? 0 : 2;   // A/B frag: lanes 0-15 K{0,1}, 16-31 K{2,3}

    // staging maps (256 threads)
    const int a_r  = tid >> 1;           // 0..127
    const int a_c  = (tid & 1) * 8;      // 0 or 8 (two b128 beats -> 8 floats)
    const int b_kp = tid >> 5;           // 0..7   K-pair this thread stages
    const int b2c  = (tid & 31) * 2;     // 0..62  column pair

    v8f acc[4] = {v8f{}, v8f{}, v8f{}, v8f{}};

    auto stage_a = [&](int buf, int k0) {
        const float* ga = A + (size_t)(m0 + a_r) * lda + (k0 + a_c);
        const uint32_t la = lds_addr_of(&sA[buf][a_r][a_c]);
        async_b128(la, ga);
        async_b128_off16(la, ga);
    };
    // B: two k-rows x two cols -> one interleaved float4 {k0c0, k1c0, k0c1, k1c1}
    auto load_b = [&](int k0) -> float4 {
        const int col = n0 + b2c;
        const float* g0 = W + (size_t)(k0 + 2 * b_kp) * ldb + col;
        const float* g1 = g0 + ldb;
        float4 v; v.x = v.y = v.z = v.w = 0.0f;
        if (col + 1 < N) { v.x = g0[0]; v.y = g1[0]; v.z = g0[1]; v.w = g1[1]; }
        else if (col < N) { v.x = g0[0]; v.y = g1[0]; }
        return v;
    };
    auto store_b = [&](int buf, float4 v) {
        *(float4*)&sBp[buf][b_kp][b2c * 2] = v;
    };

    const int nk = K / KC;
    {   // prologue: chunk 0
        float4 b0 = load_b(0);
        stage_a(0, 0);
        store_b(0, b0);
    }

    for (int i = 0; i < nk; ++i) {
        const int buf = i & 1;
        wait_async0();          // own async-A beats of chunk i landed in LDS
        __syncthreads();        // whole block's chunk i (A async + B ds) visible

        float4 bnext;
        const bool more = (i + 1 < nk);
        if (more) {
            stage_a(buf ^ 1, (i + 1) * KC);    // async, overlaps WMMAs below
            bnext = load_b((i + 1) * KC);      // loadcnt wait deferred past WMMAs
        }

        const int arow = wave * 16 + lhalf;
        #pragma unroll
        for (int kk = 0; kk < KC; kk += 4) {
            v2f a;
            a.x = sA[buf][arow][kk + khalf + 0];
            a.y = sA[buf][arow][kk + khalf + 1];
            const int kp = (kk + khalf) >> 1;
            #pragma unroll
            for (int t = 0; t < 4; ++t) {
                v2f b = *(const v2f*)&sBp[buf][kp][(t * 16 + lhalf) * 2];
                acc[t] = __builtin_amdgcn_wmma_f32_16x16x4_f32(
                             false, a, false, b, (short)0, acc[t], false, false);
            }
        }

        if (more) store_b(buf ^ 1, bnext);
        __syncthreads();        // chunk i reads done before buf is overwritten
    }

    // epilogue + store (C/D layout: VGPR r -> row r (lanes 0-15) / r+8 (16-31))
    const int row_base = m0 + wave * 16 + ((lane < 16) ? 0 : 8);
    #pragma unroll
    for (int t = 0; t < 4; ++t) {
        const int col = n0 + t * 16 + lhalf;
        if (col < N) {
            #pragma unroll
            for (int r = 0; r < 8; ++r) {
                const int row = row_base + r;
                float v = acc[t][r];
                if (EPI == 1) v = softplus_f(v + bias[col]);
                if (EPI == 2) v = mask[row] ? v : 0.0f;
                C[(size_t)row * ldc + col] = v;
            }
        }
    }
}

// ---------------------------------------------------------------------------
// Fused causal depthwise conv1d (width 4) + SiLU.  xc = xz[..., :D_INNER]
// ---------------------------------------------------------------------------
__global__ __launch_bounds__(256) void conv_silu_kernel(
    const float* __restrict__ xz, const float* __restrict__ cw,
    const float* __restrict__ cb, float* __restrict__ u)
{
    const int idx = blockIdx.x * 256 + threadIdx.x;   // (b*SEQ + l)*D_INNER + d
    const int d = idx & (D_INNER - 1);
    const int l = (idx >> 11) & (SEQ - 1);
    const int b = idx >> 21;

    float acc = cb[d];
    #pragma unroll
    for (int k = 0; k < D_CONV; ++k) {
        const int ls = l - (D_CONV - 1) + k;
        if (ls >= 0)
            acc += xz[((size_t)(b * SEQ + ls) * (2 * D_INNER)) + d] * cw[d * D_CONV + k];
    }
    u[idx] = silu_f(acc);
}

// ---------------------------------------------------------------------------
// Selective scan: one thread per (b, d); 16-wide state in registers.
// Fuses y = (scan + u*D) * silu(z).
// ---------------------------------------------------------------------------
__global__ __launch_bounds__(256) void scan_kernel(
    const float* __restrict__ delta, const float* __restrict__ u,
    const float* __restrict__ xz,    const float* __restrict__ x_dbl,
    const float* __restrict__ A_log, const float* __restrict__ Dp,
    float* __restrict__ y)
{
    const int d = blockIdx.x * 256 + threadIdx.x;     // 0..D_INNER-1
    const int b = blockIdx.y;

    float Arow[D_STATE];
    #pragma unroll
    for (int n = 0; n < D_STATE; ++n)
        Arow[n] = -__expf(A_log[d * D_STATE + n]);

    float h[D_STATE];
    #pragma unroll
    for (int n = 0; n < D_STATE; ++n) h[n] = 0.0f;

    const float Dd = Dp[d];

    for (int l = 0; l < SEQ; ++l) {
        const int rl = b * SEQ + l;
        const float dt = delta[(size_t)rl * D_INNER + d];
        const float uu = u[(size_t)rl * D_INNER + d];
        const float zz = xz[(size_t)rl * (2 * D_INNER) + D_INNER + d];
        const float du = dt * uu;
        const float* bc = x_dbl + (size_t)rl * (DT_RANK + 2 * D_STATE) + DT_RANK;

        float yacc = 0.0f;
        #pragma unroll
        for (int n = 0; n < D_STATE; ++n) {
            const float dA = __expf(dt * Arow[n]);    // v_exp_f32 (TRANS)
            h[n] = dA * h[n] + du * bc[n];            // B_t
            yacc += h[n] * bc[D_STATE + n];           // C_t
        }
        y[(size_t)rl * D_INNER + d] = (yacc + uu * Dd) * silu_f(zz);
    }
}

// ---------------------------------------------------------------------------
// Launch
// ---------------------------------------------------------------------------
extern "C" void kernel_launch(void* const* d_in, const int* in_sizes, int n_in,
                              void* d_out, int out_size, void* d_ws, size_t ws_size,
                              hipStream_t stream) {
    const float*         x         = (const float*)d_in[0];
    const unsigned char* mask      = (const unsigned char*)d_in[1];   // bool
    const float*         in_proj_w = (const float*)d_in[2];
    const float*         conv_w    = (const float*)d_in[3];
    const float*         conv_b    = (const float*)d_in[4];
    const float*         x_proj_w  = (const float*)d_in[5];
    const float*         dt_proj_w = (const float*)d_in[6];
    const float*         dt_proj_b = (const float*)d_in[7];
    const float*         A_log     = (const float*)d_in[8];
    const float*         Dp        = (const float*)d_in[9];
    const float*         out_proj_w= (const float*)d_in[10];
    float*               out       = (float*)d_out;

    // workspace carve-up (floats)
    float* ws    = (float*)d_ws;
    float* xz    = ws;                                   // ROWS * 4096
    float* u     = xz    + (size_t)ROWS * 2 * D_INNER;   // ROWS * 2048
    float* x_dbl = u     + (size_t)ROWS * D_INNER;       // ROWS * 96
    float* delta = x_dbl + (size_t)ROWS * (DT_RANK + 2 * D_STATE); // ROWS * 2048
    float* yb    = delta + (size_t)ROWS * D_INNER;       // ROWS * 2048

    dim3 blk(256);

    // 1) xz = x @ in_proj_w          (2048 x 4096 x 1024)
    wmma_gemm_f32<0><<<dim3((2 * D_INNER) / BN, ROWS / BM), blk, 0, stream>>>(
        x, in_proj_w, xz, ROWS, 2 * D_INNER, DIM, DIM, 2 * D_INNER, 2 * D_INNER,
        nullptr, nullptr);

    // 2) u = silu(causal_dwconv(xc))
    conv_silu_kernel<<<(ROWS * D_INNER) / 256, blk, 0, stream>>>(xz, conv_w, conv_b, u);

    // 3) x_dbl = u @ x_proj_w        (2048 x 96 x 2048)
    wmma_gemm_f32<0><<<dim3((DT_RANK + 2 * D_STATE + BN - 1) / BN, ROWS / BM), blk, 0, stream>>>(
        u, x_proj_w, x_dbl, ROWS, DT_RANK + 2 * D_STATE, D_INNER,
        D_INNER, DT_RANK + 2 * D_STATE, DT_RANK + 2 * D_STATE, nullptr, nullptr);

    // 4) delta = softplus(dtr @ dt_proj_w + dt_proj_b)   (2048 x 2048 x 64)
    wmma_gemm_f32<1><<<dim3(D_INNER / BN, ROWS / BM), blk, 0, stream>>>(
        x_dbl /* dtr = cols 0..63 */, dt_proj_w, delta, ROWS, D_INNER, DT_RANK,
        DT_RANK + 2 * D_STATE, D_INNER, D_INNER, dt_proj_b, nullptr);

    // 5) selective scan (fused D-residual + z-gate)
    scan_kernel<<<dim3(D_INNER / 256, B_SZ), blk, 0, stream>>>(
        delta, u, xz, x_dbl, A_log, Dp, yb);

    // 6) out = mask ? (y @ out_proj_w) : 0   (2048 x 1024 x 2048)
    wmma_gemm_f32<2><<<dim3(DIM / BN, ROWS / BM), blk, 0, stream>>>(
        yb, out_proj_w, out, ROWS, DIM, D_INNER, D_INNER, DIM, DIM,
        nullptr, mask);
}